// HybridGNNModel_52604759441817
// MI455X (gfx1250) — compile-verified
//
#include <hip/hip_runtime.h>
#include <hip/hip_bf16.h>

#define DIN   9
#define H1C   64
#define H2C   64
#define H3C   32
#define NHEAD 8
#define HGW   (NHEAD * H2C)   // 512
#define EPSBN 1e-5f

// Order-preserving encoding of -inf: bits(-inf)=0xFF800000 (negative) -> ~u
#define NEG_INF_ENC 0x007FFFFFu

typedef __attribute__((ext_vector_type(2))) float v2f;
typedef __attribute__((ext_vector_type(8))) float v8f;

// ---------- order-preserving float<->uint for atomicMax segment-max ----------
__device__ __forceinline__ unsigned fenc(float f) {
  unsigned u = __float_as_uint(f);
  return (u & 0x80000000u) ? ~u : (u | 0x80000000u);
}
__device__ __forceinline__ float fdec(unsigned e) {
  unsigned u = (e & 0x80000000u) ? (e ^ 0x80000000u) : ~e;
  return __uint_as_float(u);
}

// ------------------------------- fills ---------------------------------------
__global__ void k_fill_f32(float* p, float v, long long n) {
  long long i = (long long)blockIdx.x * blockDim.x + threadIdx.x;
  long long s = (long long)gridDim.x * blockDim.x;
  for (; i < n; i += s) p[i] = v;
}
__global__ void k_fill_u32(unsigned* p, unsigned v, long long n) {
  long long i = (long long)blockIdx.x * blockDim.x + threadIdx.x;
  long long s = (long long)gridDim.x * blockDim.x;
  for (; i < n; i += s) p[i] = v;
}

// --------------------------- degree / norm -----------------------------------
__global__ void k_deg_edges(const int* __restrict__ col, int E, float* __restrict__ deg) {
  int i = blockIdx.x * blockDim.x + threadIdx.x;
  if (i < E) atomicAdd(&deg[col[i]], 1.0f);
}
__global__ void k_deg_to_dis(float* __restrict__ deg, int N) {
  int i = blockIdx.x * blockDim.x + threadIdx.x;
  if (i < N) {
    float d = deg[i];
    d = d < 1.0f ? 1.0f : d;
    deg[i] = rsqrtf(d);
  }
}

// ----------------------- GCN1 input linear (K=9, tiny) -----------------------
__global__ void k_lin_x_w1(const float* __restrict__ x, const float* __restrict__ W1,
                           float* __restrict__ h, int N) {
  int idx = blockIdx.x * blockDim.x + threadIdx.x;
  if (idx >= N * H1C) return;
  int n = idx / H1C, c = idx % H1C;
  float s = 0.0f;
#pragma unroll
  for (int k = 0; k < DIN; ++k) s += x[n * DIN + k] * W1[k * H1C + c];
  h[idx] = s;
}

// -------------------- GCN scatter: out[col] += h[row]*norm -------------------
__global__ void k_gcn_agg(const float* __restrict__ h, const float* __restrict__ dis,
                          const int* __restrict__ row, const int* __restrict__ col,
                          int E, int N, int C, float* __restrict__ out) {
  int t = blockIdx.x * blockDim.x + threadIdx.x;
  if (t >= E + N) return;
  int r, c;
  if (t < E) { r = row[t]; c = col[t]; } else { r = c = t - E; }
  float nrm = dis[r] * dis[c];
  const float* hr = h + (long long)r * C;
  float* oc = out + (long long)c * C;
  for (int k = 0; k < C; ++k) atomicAdd(&oc[k], hr[k] * nrm);
}

__global__ void k_add_bias(float* __restrict__ h, const float* __restrict__ b,
                           long long total, int C) {
  long long i = (long long)blockIdx.x * blockDim.x + threadIdx.x;
  if (i < total) h[i] += b[i % C];
}

// ------------------------- BatchNorm (training mode) -------------------------
__global__ void k_bn_stats(const float* __restrict__ h, long long total, int C,
                           float* __restrict__ sums, float* __restrict__ sqs) {
  __shared__ float ssum[64];
  __shared__ float ssq[64];
  for (int c = threadIdx.x; c < C; c += blockDim.x) { ssum[c] = 0.f; ssq[c] = 0.f; }
  __syncthreads();
  long long i = (long long)blockIdx.x * blockDim.x + threadIdx.x;
  long long s = (long long)gridDim.x * blockDim.x;
  for (; i < total; i += s) {
    float v = h[i];
    int c = (int)(i % C);
    atomicAdd(&ssum[c], v);      // ds_add_f32
    atomicAdd(&ssq[c], v * v);
  }
  __syncthreads();
  for (int c = threadIdx.x; c < C; c += blockDim.x) {
    atomicAdd(&sums[c], ssum[c]);
    atomicAdd(&sqs[c], ssq[c]);
  }
}
__global__ void k_bn_finalize(const float* __restrict__ sums, const float* __restrict__ sqs,
                              const float* __restrict__ g, const float* __restrict__ be,
                              int N, int C, float* __restrict__ scl, float* __restrict__ sft) {
  int c = blockIdx.x * blockDim.x + threadIdx.x;
  if (c >= C) return;
  float invN = 1.0f / (float)N;
  float mu = sums[c] * invN;
  float var = sqs[c] * invN - mu * mu;
  float sc = g[c] * rsqrtf(var + EPSBN);
  scl[c] = sc;
  sft[c] = be[c] - mu * sc;
}
// act: 0 = ReLU, 1 = ELU(alpha=1)
__global__ void k_bn_apply(float* __restrict__ h, const float* __restrict__ scl,
                           const float* __restrict__ sft, long long total, int C, int act) {
  long long i = (long long)blockIdx.x * blockDim.x + threadIdx.x;
  if (i >= total) return;
  int c = (int)(i % C);
  float v = h[i] * scl[c] + sft[c];
  if (act == 0) v = fmaxf(v, 0.0f);
  else          v = v > 0.0f ? v : (__expf(v) - 1.0f);
  h[i] = v;
}

// --------------- fp32 WMMA GEMM: C[M,Nc] = A[M,K] @ B[K,Nc] ------------------
// One wave per 16x16 tile; K stepped by 4 via V_WMMA_F32_16X16X4_F32.
// Fragment layouts per CDNA5 ISA 7.12.2 (wave32).
// Bases computed once; inner loop uses pure pointer bumps (no per-iter 64-bit mul).
__global__ void k_gemm_wmma_f32(const float* __restrict__ A, const float* __restrict__ B,
                                float* __restrict__ C, int M, int Nc, int K) {
  int wave = blockIdx.x * (blockDim.x >> 5) + (threadIdx.x >> 5);
  int nT = Nc >> 4;
  int mT = (M + 15) >> 4;
  if (wave >= mT * nT) return;
  int mB = (wave / nT) << 4;
  int nB = (wave % nT) << 4;
  int lane = threadIdx.x & 31;
  int half = lane >> 4;   // 0: low half (K+0/1), 1: high half (K+2/3)
  int r    = lane & 15;

  int ar = mB + r;
  if (ar >= M) ar = M - 1;                 // clamp (M is 16-aligned here anyway)

  const float* Ap = A + (size_t)ar * K + (half << 1);
  const float* Bp = B + (size_t)(half << 1) * Nc + nB + r;
  const size_t bStep = (size_t)Nc * 4;     // K advances by 4 rows per iter

  v8f acc = {0.f, 0.f, 0.f, 0.f, 0.f, 0.f, 0.f, 0.f};
  for (int k = 0; k < K; k += 4) {
    v2f a, b;
    a[0] = Ap[0];
    a[1] = Ap[1];
    b[0] = Bp[0];
    b[1] = Bp[Nc];
    acc = __builtin_amdgcn_wmma_f32_16x16x4_f32(
        /*neg_a=*/false, a, /*neg_b=*/false, b,
        /*c_mod=*/(short)0, acc, /*reuse_a=*/false, /*reuse_b=*/false);
    Ap += 4;
    Bp += bStep;
  }

  float* Cp = C + (size_t)(mB + (half << 3)) * Nc + nB + r;
#pragma unroll
  for (int j = 0; j < 8; ++j) {
    int rowIdx = mB + j + (half << 3);     // C VGPR j -> M=j (low) / M=j+8 (high)
    if (rowIdx < M) Cp[(size_t)j * Nc] = acc[j];
  }
}

// ------------------------------- GAT -----------------------------------------
__global__ void k_gat_scores(const float* __restrict__ hg, const float* __restrict__ a_src,
                             const float* __restrict__ a_dst, int N,
                             float* __restrict__ asrc, float* __restrict__ adst) {
  int idx = blockIdx.x * blockDim.x + threadIdx.x;
  if (idx >= N * NHEAD) return;
  int n = idx / NHEAD, h = idx % NHEAD;
  const float* v  = hg + (long long)n * HGW + h * H2C;
  const float* as = a_src + h * H2C;
  const float* ad = a_dst + h * H2C;
  float s0 = 0.f, s1 = 0.f;
#pragma unroll 8
  for (int k = 0; k < H2C; ++k) { float x = v[k]; s0 += x * as[k]; s1 += x * ad[k]; }
  asrc[idx] = s0;
  adst[idx] = s1;
}

__device__ __forceinline__ void edge_rc(const int* row, const int* col, int E, int t,
                                        int& r, int& c) {
  if (t < E) { r = row[t]; c = col[t]; } else { r = c = t - E; }
}
__device__ __forceinline__ float lrelu02(float v) { return v > 0.f ? v : 0.2f * v; }

__global__ void k_gat_max(const float* __restrict__ asrc, const float* __restrict__ adst,
                          const int* __restrict__ row, const int* __restrict__ col,
                          int E, int N, unsigned* __restrict__ mmax) {
  int t = blockIdx.x * blockDim.x + threadIdx.x;
  if (t >= (E + N) * NHEAD) return;
  int e = t / NHEAD, h = t % NHEAD;
  int r, c; edge_rc(row, col, E, e, r, c);
  float v = lrelu02(asrc[r * NHEAD + h] + adst[c * NHEAD + h]);
  atomicMax(&mmax[c * NHEAD + h], fenc(v));
}

__global__ void k_gat_denom(const float* __restrict__ asrc, const float* __restrict__ adst,
                            const unsigned* __restrict__ mmax,
                            const int* __restrict__ row, const int* __restrict__ col,
                            int E, int N, float* __restrict__ denom) {
  int t = blockIdx.x * blockDim.x + threadIdx.x;
  if (t >= (E + N) * NHEAD) return;
  int e = t / NHEAD, h = t % NHEAD;
  int r, c; edge_rc(row, col, E, e, r, c);
  float v = lrelu02(asrc[r * NHEAD + h] + adst[c * NHEAD + h]);
  float ex = __expf(v - fdec(mmax[c * NHEAD + h]));
  atomicAdd(&denom[c * NHEAD + h], ex);
}

// out[col, :] += (alpha / NHEAD) * hg[row, h, :]   (head-mean folded in)
__global__ void k_gat_agg(const float* __restrict__ asrc, const float* __restrict__ adst,
                          const unsigned* __restrict__ mmax, const float* __restrict__ denom,
                          const float* __restrict__ hg,
                          const int* __restrict__ row, const int* __restrict__ col,
                          int E, int N, float* __restrict__ out) {
  int t = blockIdx.x * blockDim.x + threadIdx.x;
  if (t >= (E + N) * NHEAD) return;
  int e = t / NHEAD, h = t % NHEAD;
  int r, c; edge_rc(row, col, E, e, r, c);
  float v = lrelu02(asrc[r * NHEAD + h] + adst[c * NHEAD + h]);
  float alpha = __expf(v - fdec(mmax[c * NHEAD + h])) / denom[c * NHEAD + h];
  alpha *= (1.0f / NHEAD);
  const float* src = hg + (long long)r * HGW + h * H2C;
  float* dst = out + (long long)c * H2C;
  for (int k = 0; k < H2C; ++k) atomicAdd(&dst[k], alpha * src[k]);
}

// ------------------------------ FC head --------------------------------------
__global__ void k_fc(const float* __restrict__ h3, const float* __restrict__ Wf,
                     const float* __restrict__ bf, float* __restrict__ out, int N) {
  int n = blockIdx.x * blockDim.x + threadIdx.x;
  if (n >= N) return;
  float s = bf[0];
#pragma unroll
  for (int k = 0; k < H3C; ++k) s += h3[(long long)n * H3C + k] * Wf[k];
  out[n] = 1.0f / (1.0f + __expf(-s));
}

// =============================================================================
extern "C" void kernel_launch(void* const* d_in, const int* in_sizes, int n_in,
                              void* d_out, int out_size, void* d_ws, size_t ws_size,
                              hipStream_t stream) {
  const float* x     = (const float*)d_in[0];
  const int*   ei    = (const int*)d_in[1];
  const float* W1    = (const float*)d_in[2];
  const float* b1    = (const float*)d_in[3];
  const float* g1    = (const float*)d_in[4];
  const float* be1   = (const float*)d_in[5];
  const float* Wg    = (const float*)d_in[6];
  const float* a_src = (const float*)d_in[7];
  const float* a_dst = (const float*)d_in[8];
  const float* bg    = (const float*)d_in[9];
  const float* g2    = (const float*)d_in[10];
  const float* be2   = (const float*)d_in[11];
  const float* W2    = (const float*)d_in[12];
  const float* b2    = (const float*)d_in[13];
  const float* g3    = (const float*)d_in[14];
  const float* be3   = (const float*)d_in[15];
  const float* Wf    = (const float*)d_in[16];
  const float* bf    = (const float*)d_in[17];

  const int N = in_sizes[0] / DIN;
  const int E = in_sizes[1] / 2;
  const int* row = ei;
  const int* col = ei + E;
  float* out = (float*)d_out;

  // ---- carve workspace ----
  char* wsp = (char*)d_ws;
  auto carve = [&](size_t bytes) -> void* {
    void* p = wsp;
    wsp += (bytes + 255) & ~(size_t)255;
    return p;
  };
  float*    dis   = (float*)carve((size_t)N * 4);                 // deg -> 1/sqrt(deg)
  float*    h1    = (float*)carve((size_t)N * H1C * 4);           // GCN1 out (also GCN2 pre-agg reuse)
  float*    hg    = (float*)carve((size_t)N * HGW * 4);           // GAT per-head features
  float*    asrc  = (float*)carve((size_t)N * NHEAD * 4);
  float*    adst  = (float*)carve((size_t)N * NHEAD * 4);
  unsigned* mmax  = (unsigned*)carve((size_t)N * NHEAD * 4);
  float*    denom = (float*)carve((size_t)N * NHEAD * 4);
  float*    h2    = (float*)carve((size_t)N * H2C * 4);           // GAT out (also GCN1 pre-agg reuse)
  float*    h3    = (float*)carve((size_t)N * H3C * 4);
  float*    sums  = (float*)carve(64 * 4);
  float*    sqs   = (float*)carve(64 * 4);
  float*    scl   = (float*)carve(64 * 4);
  float*    sft   = (float*)carve(64 * 4);

  const int TB = 256;
  auto cdiv = [](long long a, long long b) { return (int)((a + b - 1) / b); };
  const int Et  = E + N;
  const int EtH = Et * NHEAD;

  // ---- degrees + symmetric norm ----
  k_fill_f32<<<cdiv(N, TB), TB, 0, stream>>>(dis, 1.0f, N);       // self-loop contribution
  k_deg_edges<<<cdiv(E, TB), TB, 0, stream>>>(col, E, dis);
  k_deg_to_dis<<<cdiv(N, TB), TB, 0, stream>>>(dis, N);

  // ---- GCN1: pre-agg (into h2 as scratch), scatter, bias, BN+ReLU ----
  k_lin_x_w1<<<cdiv((long long)N * H1C, TB), TB, 0, stream>>>(x, W1, h2, N);
  k_fill_f32<<<cdiv((long long)N * H1C, TB), TB, 0, stream>>>(h1, 0.0f, (long long)N * H1C);
  k_gcn_agg<<<cdiv(Et, TB), TB, 0, stream>>>(h2, dis, row, col, E, N, H1C, h1);
  k_add_bias<<<cdiv((long long)N * H1C, TB), TB, 0, stream>>>(h1, b1, (long long)N * H1C, H1C);
  k_fill_f32<<<1, TB, 0, stream>>>(sums, 0.0f, 64);
  k_fill_f32<<<1, TB, 0, stream>>>(sqs, 0.0f, 64);
  k_bn_stats<<<512, TB, 0, stream>>>(h1, (long long)N * H1C, H1C, sums, sqs);
  k_bn_finalize<<<1, 64, 0, stream>>>(sums, sqs, g1, be1, N, H1C, scl, sft);
  k_bn_apply<<<cdiv((long long)N * H1C, TB), TB, 0, stream>>>(h1, scl, sft, (long long)N * H1C, H1C, 0);

  // ---- GAT: WMMA GEMM h1[N,64] @ Wg[64,512] -> hg ----
  {
    int tiles = ((N + 15) / 16) * (HGW / 16);
    k_gemm_wmma_f32<<<cdiv((long long)tiles * 32, TB), TB, 0, stream>>>(h1, Wg, hg, N, HGW, H1C);
  }
  k_gat_scores<<<cdiv((long long)N * NHEAD, TB), TB, 0, stream>>>(hg, a_src, a_dst, N, asrc, adst);
  k_fill_u32<<<cdiv((long long)N * NHEAD, TB), TB, 0, stream>>>(mmax, NEG_INF_ENC, (long long)N * NHEAD);
  k_gat_max<<<cdiv(EtH, TB), TB, 0, stream>>>(asrc, adst, row, col, E, N, mmax);
  k_fill_f32<<<cdiv((long long)N * NHEAD, TB), TB, 0, stream>>>(denom, 0.0f, (long long)N * NHEAD);
  k_gat_denom<<<cdiv(EtH, TB), TB, 0, stream>>>(asrc, adst, mmax, row, col, E, N, denom);
  k_fill_f32<<<cdiv((long long)N * H2C, TB), TB, 0, stream>>>(h2, 0.0f, (long long)N * H2C);
  k_gat_agg<<<cdiv(EtH, TB), TB, 0, stream>>>(asrc, adst, mmax, denom, hg, row, col, E, N, h2);
  k_add_bias<<<cdiv((long long)N * H2C, TB), TB, 0, stream>>>(h2, bg, (long long)N * H2C, H2C);
  k_fill_f32<<<1, TB, 0, stream>>>(sums, 0.0f, 64);
  k_fill_f32<<<1, TB, 0, stream>>>(sqs, 0.0f, 64);
  k_bn_stats<<<512, TB, 0, stream>>>(h2, (long long)N * H2C, H2C, sums, sqs);
  k_bn_finalize<<<1, 64, 0, stream>>>(sums, sqs, g2, be2, N, H2C, scl, sft);
  k_bn_apply<<<cdiv((long long)N * H2C, TB), TB, 0, stream>>>(h2, scl, sft, (long long)N * H2C, H2C, 1);

  // ---- GCN2: WMMA GEMM h2[N,64] @ W2[64,32] -> h1 (scratch), scatter, BN+ReLU ----
  {
    int tiles = ((N + 15) / 16) * (H3C / 16);
    k_gemm_wmma_f32<<<cdiv((long long)tiles * 32, TB), TB, 0, stream>>>(h2, W2, h1, N, H3C, H2C);
  }
  k_fill_f32<<<cdiv((long long)N * H3C, TB), TB, 0, stream>>>(h3, 0.0f, (long long)N * H3C);
  k_gcn_agg<<<cdiv(Et, TB), TB, 0, stream>>>(h1, dis, row, col, E, N, H3C, h3);
  k_add_bias<<<cdiv((long long)N * H3C, TB), TB, 0, stream>>>(h3, b2, (long long)N * H3C, H3C);
  k_fill_f32<<<1, TB, 0, stream>>>(sums, 0.0f, 64);
  k_fill_f32<<<1, TB, 0, stream>>>(sqs, 0.0f, 64);
  k_bn_stats<<<512, TB, 0, stream>>>(h3, (long long)N * H3C, H3C, sums, sqs);
  k_bn_finalize<<<1, 64, 0, stream>>>(sums, sqs, g3, be3, N, H3C, scl, sft);
  k_bn_apply<<<cdiv((long long)N * H3C, TB), TB, 0, stream>>>(h3, scl, sft, (long long)N * H3C, H3C, 0);

  // ---- FC + sigmoid ----
  k_fc<<<cdiv(N, TB), TB, 0, stream>>>(h3, Wf, bf, out, N);
}